// AngularMaxPooling_74749610819722
// MI455X (gfx1250) — compile-verified
//
#include <hip/hip_runtime.h>

// Angular max pooling: inputs [B=8, R=16, N=6144, F=128] f32 -> out [B, N, F].
// Pure bandwidth problem (~428 MB total => ~18us at 23.3 TB/s roofline).
// One wave32 per (b, n) group:
//   1) 16 rows staged global->LDS via GLOBAL_LOAD_ASYNC_TO_LDS_B128 (ASYNCcnt,
//      no VGPR round-trip),
//   2) rotation sum-of-squares = diag(A*A^T) via V_WMMA_F32_16X16X4_F32,
//   3) wave butterfly argmax (first-index tie break = jnp.argmax),
//   4) winner row emitted straight from LDS (input bytes touch HBM exactly once).

typedef __attribute__((ext_vector_type(2))) float v2f;
typedef __attribute__((ext_vector_type(8))) float v8f;

#define B_ 8
#define R_ 16
#define N_ 6144
#define F_ 128
#define ROWSTRIDE 132              // 128 + 4 dword pad: conflict-free strided LDS reads
#define WAVES_PER_BLOCK 8
#define PLANE_BYTES (N_ * F_ * 4)  // bytes between rotation planes: 3,145,728

__global__ __launch_bounds__(256) void ang_max_pool_wmma(const float* __restrict__ in,
                                                         float* __restrict__ out) {
    __shared__ float tile[WAVES_PER_BLOCK][R_ * ROWSTRIDE];
    const int lane = threadIdx.x & 31;
    const int w    = threadIdx.x >> 5;
    const int g    = blockIdx.x * WAVES_PER_BLOCK + w;   // group id in [0, B*N)
    const int b    = g / N_;
    const int n    = g % N_;

    // ---- stage the 16 rotation rows (512B each) straight into LDS ----
    // GVS addressing: SGPR batch-plane base + 32-bit per-lane offset (max ~47MB).
    // LDS dest = low 32 bits of the generic pointer to the shared tile.
    const float* gbase = in + (size_t)b * R_ * N_ * F_;              // wave-uniform -> SGPR
    const unsigned lds0  = (unsigned)(uintptr_t)(&tile[w][0]) + (unsigned)(lane * 16);
    const unsigned gofs0 = (unsigned)(n * (F_ * 4) + lane * 16);
    #pragma unroll
    for (int r = 0; r < R_; ++r) {
        const unsigned ldsa = lds0 + (unsigned)(r * (ROWSTRIDE * 4));
        const unsigned gofs = gofs0 + (unsigned)(r * PLANE_BYTES);
        asm volatile("global_load_async_to_lds_b128 %0, %1, %2"
                     :: "v"(ldsa), "v"(gofs), "s"(gbase) : "memory");
    }
    asm volatile("s_wait_asynccnt 0" ::: "memory");   // LDS tile complete & visible

    // ---- Gram matrix via WMMA: D = A * A^T ; diag(D) = per-rotation sum-of-squares.
    // f32 16x16x4 A layout: lane L holds M = L%16, K = 2*(L/16) + {0,1}  -> one float2.
    // The B 4x16 layout is the mirrored striping, so the SAME register pair is A and B.
    v8f acc = {};
    #pragma unroll
    for (int kc = 0; kc < F_ / 4; ++kc) {
        const int col = kc * 4 + 2 * (lane >> 4);
        v2f a = *(const v2f*)&tile[w][(lane & 15) * ROWSTRIDE + col];
        acc = __builtin_amdgcn_wmma_f32_16x16x4_f32(false, a, false, a,
                                                    (short)0, acc, false, false);
    }

    // ---- pull diag(D): r<8 -> (vgpr r, lane r); r>=8 -> (vgpr r-8, lane r+16) ----
    const bool active = (lane < 8) || (lane >= 24);
    const int  vsel   = (lane < 8) ? lane : lane - 24;
    float val = -__builtin_inff();
    #pragma unroll
    for (int i = 0; i < 8; ++i)
        if (vsel == i) val = acc[i];            // short v_cndmask chain
    int ridx = (lane < 8) ? lane : lane - 16;
    if (!active) { val = -__builtin_inff(); ridx = R_; }

    // ---- wave-wide argmax with first-index tie break (jnp.argmax semantics) ----
    #pragma unroll
    for (int off = 16; off > 0; off >>= 1) {
        const float ov = __shfl_xor(val, off, 32);
        const int   oi = __shfl_xor(ridx, off, 32);
        if (ov > val || (ov == val && oi < ridx)) { val = ov; ridx = oi; }
    }

    // ---- emit the winning rotation's row straight from LDS (no HBM re-read) ----
    const float4 wv = *(const float4*)&tile[w][ridx * ROWSTRIDE + lane * 4];
    *(float4*)(out + ((size_t)b * N_ + (size_t)n) * F_ + lane * 4) = wv;
}

extern "C" void kernel_launch(void* const* d_in, const int* in_sizes, int n_in,
                              void* d_out, int out_size, void* d_ws, size_t ws_size,
                              hipStream_t stream) {
    const float* in  = (const float*)d_in[0];
    float*       out = (float*)d_out;
    const int groups = B_ * N_;                       // 49152 waves total
    dim3 block(32 * WAVES_PER_BLOCK);                 // 256 threads = 8 waves
    dim3 grid(groups / WAVES_PER_BLOCK);              // 6144 blocks (exact cover)
    ang_max_pool_wmma<<<grid, block, 0, stream>>>(in, out);
}